// Spatial_Attention_19404662243735
// MI455X (gfx1250) — compile-verified
//
#include <hip/hip_runtime.h>
#include <cstdint>

// ---------------------------------------------------------------------------
// Spatial attention: pool(mean/max over C) -> 7^3 conv (2->1 ch) -> sigmoid*x
// x: [2, 32, 64, 128, 128] f32, W: [1, 2, 7, 7, 7] f32
// ---------------------------------------------------------------------------

typedef float v2f  __attribute__((ext_vector_type(2)));
typedef float v8f  __attribute__((ext_vector_type(8)));
typedef unsigned int u32x4 __attribute__((ext_vector_type(4)));
typedef int  i32x4 __attribute__((ext_vector_type(4)));
typedef int  i32x8 __attribute__((ext_vector_type(8)));

#define NB      2
#define NC      32
#define ND      64
#define NH      128
#define NW      128
#define PADW    134              // 128 + 2*3
#define PADH    134
#define PADD    70               // 64 + 2*3
#define PPLANE  (PADW * PADH)    // 17956
#define PVOL    (PPLANE * PADD)  // 1256920 floats per (b, c) volume
#define PAD_FLOATS (NB * 2 * PVOL)         // padded pooled tensor [2][2][70][134][134]
#define SPATIAL4   (ND * NH * NW / 4)      // 262144 float4 per (b, c) plane

// ---------------------------------------------------------------------------
// Pass 1: channel mean + max -> padded x1 buffer (halo pre-zeroed by memset)
// ---------------------------------------------------------------------------
__global__ void __launch_bounds__(256)
pool_kernel(const float4* __restrict__ x4, float* __restrict__ pad) {
    unsigned tid = blockIdx.x * blockDim.x + threadIdx.x;   // 524288 threads
    unsigned w4 = tid & 31;  unsigned r = tid >> 5;
    unsigned h  = r & 127;   r >>= 7;
    unsigned z  = r & 63;    unsigned b = r >> 6;
    unsigned sp = (z * 128u + h) * 32u + w4;

    float4 s  = make_float4(0.f, 0.f, 0.f, 0.f);
    float4 mx = make_float4(-3.402823466e38f, -3.402823466e38f,
                            -3.402823466e38f, -3.402823466e38f);
    for (int c = 0; c < NC; ++c) {
        float4 v = x4[(b * NC + (unsigned)c) * (unsigned)SPATIAL4 + sp];
        s.x += v.x; s.y += v.y; s.z += v.z; s.w += v.w;
        mx.x = fmaxf(mx.x, v.x); mx.y = fmaxf(mx.y, v.y);
        mx.z = fmaxf(mx.z, v.z); mx.w = fmaxf(mx.w, v.w);
    }
    const float inv = 1.0f / 32.0f;
    // padded layout [b][c01][z+3][h+3][w+3]
    unsigned pb = (((b * 2u + 0u) * PADD + z + 3u) * PADH + h + 3u) * PADW + w4 * 4u + 3u;
    pad[pb + 0] = s.x * inv;  pad[pb + 1] = s.y * inv;
    pad[pb + 2] = s.z * inv;  pad[pb + 3] = s.w * inv;
    pad[pb + PVOL + 0] = mx.x; pad[pb + PVOL + 1] = mx.y;
    pad[pb + PVOL + 2] = mx.z; pad[pb + PVOL + 3] = mx.w;
}

// ---------------------------------------------------------------------------
// Pass 2: 7x7x7 conv via WMMA Toeplitz GEMM + fused sigmoid.
// One workgroup (4 waves) = 32x32 output tile at one (b, z).
// Input patch (2c x 7z x 38h x 40w f32, 85 KB) staged into LDS by one TDM op.
// Per (c,dz,dy): out[16x16] += A[16x24] x T[24x16], 6x v_wmma_f32_16x16x4_f32.
//   A[m][k]  = patch row value at (h0+sh+m+dy, w0+sw+k)   (ds_load_b64)
//   T[k][n]  = w7[k-n] if 0<=k-n<7 else 0                 (built from LDS wts)
// ---------------------------------------------------------------------------
__global__ void __launch_bounds__(128)
conv_wmma_kernel(const float* __restrict__ Wg, const float* __restrict__ pad,
                 float* __restrict__ smap) {
    __shared__ float tile[2 * 7 * 38 * 40];   // 21280 floats = 85120 B
    __shared__ float wlds[2 * 7 * 7 * 8];     // taps padded to 8 per (c,dz,dy)

    const int tid  = threadIdx.x;
    const int lane = tid & 31;
    const int wid  = tid >> 5;
    const int bz   = blockIdx.z;              // b*64 + z
    const int h0   = blockIdx.y * 32;
    const int w0   = blockIdx.x * 32;
    const int b    = bz >> 6;
    const int z0   = bz & 63;

    // stage compact weights [c][dz][dy][dx(pad8)]
    for (int i = tid; i < 686; i += 128)
        wlds[(i / 7) * 8 + (i % 7)] = Wg[i];

    // wave 0 issues one Tensor-Data-Mover load of the whole input patch
    if (wid == 0) {
        unsigned long long gaddr = (unsigned long long)(uintptr_t)pad +
            4ull * ((((unsigned long long)(b * 2) * PADD + (unsigned)z0) * PADH +
                     (unsigned)h0) * PADW + (unsigned)w0);
        unsigned int lds_addr = (unsigned int)(uintptr_t)&tile[0];
        u32x4 g0; i32x8 g1; i32x4 g2, g3; i32x8 g4;
        // D# group 0: count=1 | lds_addr | global_addr | type=2
        g0.x = 1u;
        g0.y = lds_addr;
        g0.z = (unsigned int)(gaddr & 0xFFFFFFFFull);
        g0.w = (unsigned int)((gaddr >> 32) & 0x1FFFFFFull) | (2u << 30);
        // D# group 1
        g1[0] = (2 << 16);                // workgroup_mask=0, data_size=4B
        g1[1] = (PADW << 16);             // tensor_dim0 lo16 (bits 63:48)
        g1[2] = (PADH << 16);             // dim0 hi=0 | tensor_dim1 lo16
        g1[3] = (40 << 16);               // dim1 hi=0 | tile_dim0 = 40 (w)
        g1[4] = 38 | (7 << 16);           // tile_dim1 = 38 (h) | tile_dim2 = 7 (z)
        g1[5] = PADW;                     // tensor_dim0_stride (row pitch)
        g1[6] = (PPLANE << 16);           // stride0 hi=0 | tensor_dim1_stride lo16
        g1[7] = 0;                        // tensor_dim1_stride hi
        // D# group 2
        g2[0] = PADD;                     // tensor_dim2 (z extent)
        g2[1] = 2;                        // tensor_dim3 (channels)
        g2[2] = PVOL;                     // tensor_dim2_stride (c pitch)
        g2[3] = (2 << 16);                // stride2 hi=0 | tile_dim3 = 2 (c)
        // D# group 3 (dim4 unused)
        g3[0] = 0; g3[1] = 0; g3[2] = 0; g3[3] = 0;
        // extra descriptor group (6-arg toolchain variant): zero-filled
        g4[0] = 0; g4[1] = 0; g4[2] = 0; g4[3] = 0;
        g4[4] = 0; g4[5] = 0; g4[6] = 0; g4[7] = 0;
        __builtin_amdgcn_tensor_load_to_lds(g0, g1, g2, g3, g4, 0);
        __builtin_amdgcn_s_wait_tensorcnt(0);
    }
    __syncthreads();

    const int sh = (wid >> 1) * 16;       // 16x16 sub-tile per wave
    const int sw = (wid & 1) * 16;
    const int m  = lane & 15;             // A-matrix row / D-matrix N lane
    const int n  = lane & 15;
    const int ko = (lane >> 4) * 2;       // A/B K split: lanes16-31 hold K+2

    v8f acc = {0.f, 0.f, 0.f, 0.f, 0.f, 0.f, 0.f, 0.f};

    for (int c = 0; c < 2; ++c)
        for (int dz = 0; dz < 7; ++dz)
            for (int dy = 0; dy < 7; ++dy) {
                const int cb = ((c * 7 + dz) * 7 + dy) * 8;
                const float* arow =
                    &tile[(((c * 7 + dz) * 38) + sh + dy + m) * 40 + sw];
#pragma unroll
                for (int j = 0; j < 6; ++j) {
                    const int k0 = 4 * j + ko;
                    // A chunk: X[m][k0], X[m][k0+1] -> one ds_load_b64
                    v2f a = *reinterpret_cast<const v2f*>(arow + k0);
                    // Toeplitz B chunk: T[k][n] = w7[k-n] inside the band
                    int dx0 = k0 - n;
                    int dx1 = dx0 + 1;
                    int c0 = dx0 < 0 ? 0 : (dx0 > 6 ? 6 : dx0);
                    int c1 = dx1 < 0 ? 0 : (dx1 > 6 ? 6 : dx1);
                    float t0 = wlds[cb + c0];
                    float t1 = wlds[cb + c1];
                    v2f bb;
                    bb.x = (c0 == dx0) ? t0 : 0.f;
                    bb.y = (c1 == dx1) ? t1 : 0.f;
                    acc = __builtin_amdgcn_wmma_f32_16x16x4_f32(
                        false, a, false, bb, (short)0, acc, false, false);
                }
            }

    // C/D layout: VGPR r -> M = r + 8*lane[4], N = lane[3:0]; fuse sigmoid
    const int mbase = (lane >> 4) * 8;
#pragma unroll
    for (int r = 0; r < 8; ++r) {
        int hh = h0 + sh + mbase + r;
        int ww = w0 + sw + n;
        float v = acc[r];
        smap[(bz * NH + hh) * NW + ww] = 1.0f / (1.0f + __expf(-v));
    }
}

// ---------------------------------------------------------------------------
// Pass 3: out = sigmoid_map * x, broadcast over 32 channels (pure bandwidth)
// ---------------------------------------------------------------------------
__global__ void __launch_bounds__(256)
gate_kernel(const float4* __restrict__ x4, const float4* __restrict__ s4,
            float4* __restrict__ o4) {
    unsigned tid = blockIdx.x * blockDim.x + threadIdx.x;  // 16777216 threads
    unsigned sp4 = tid & (SPATIAL4 - 1);
    unsigned bc  = tid >> 18;          // (b*32 + c)
    unsigned b   = bc >> 5;
    float4 s = s4[b * (unsigned)SPATIAL4 + sp4];
    float4 v = x4[tid];
    float4 o;
    o.x = v.x * s.x; o.y = v.y * s.y; o.z = v.z * s.z; o.w = v.w * s.w;
    o4[tid] = o;
}

// ---------------------------------------------------------------------------
extern "C" void kernel_launch(void* const* d_in, const int* in_sizes, int n_in,
                              void* d_out, int out_size, void* d_ws, size_t ws_size,
                              hipStream_t stream) {
    const float* x  = (const float*)d_in[0];   // [2,32,64,128,128]
    const float* Wg = (const float*)d_in[1];   // [1,2,7,7,7] = 686 floats
    float* out  = (float*)d_out;

    float* pad  = (float*)d_ws;                // padded pooled x1, 20.1 MB
    float* smap = pad + PAD_FLOATS;            // sigmoid map, 8.4 MB

    // zero the halo (and interior for determinism), then the three passes
    (void)hipMemsetAsync(pad, 0, (size_t)PAD_FLOATS * sizeof(float), stream);

    pool_kernel<<<2048, 256, 0, stream>>>((const float4*)x, pad);

    dim3 cgrid(4, 4, NB * ND);                 // (w tiles, h tiles, b*z)
    conv_wmma_kernel<<<cgrid, 128, 0, stream>>>(Wg, pad, smap);

    gate_kernel<<<65536, 256, 0, stream>>>((const float4*)x,
                                           (const float4*)smap, (float4*)out);
}